// GATorderedIMP4_88175678587739
// MI455X (gfx1250) — compile-verified
//
#include <hip/hip_runtime.h>
#include <hip/hip_bf16.h>
#include <cmath>

// ---------------------------------------------------------------------------
// GAT (2-layer, multi-head) for MI455X / gfx1250.
// Dense projections use V_WMMA_F32_16X16X4_F32 (full fp32 precision; the
// workload is HBM/atomic bound, so low-precision WMMA buys nothing).
// Weights are pre-packed into a K-pair-interleaved padded layout so the
// fully-unrolled K loop is pure {global_load_b64 x2, v_wmma} with immediate
// offsets -- no guards, no 64-bit address math in the hot loop.
// ---------------------------------------------------------------------------

typedef __attribute__((ext_vector_type(2))) float v2f;
typedef __attribute__((ext_vector_type(8))) float v8f;

#define GAT_N     50000
#define GAT_E     800000
#define GAT_FIN   128
#define GAT_FH    32
#define GAT_H     4
#define GAT_C1    128   // H * FH
#define GAT_NC    40
#define GAT_NCP   48    // NC padded to col-tile multiple
#define GAT_ALPHA 0.2f

// ---------------- WMMA fp32 GEMM: C[M,NCSTORE] = A[M,K] * B[K,NCSTORE] -----
// One wave (32 lanes) computes a 16x16 tile of C.
// Per ISA 7.12.2 (wave32):
//   A 16x4  : 2 VGPRs. lanes 0-15 -> M=lane, K=(k0,k0+1); lanes 16-31 -> M=lane-16, K=(k0+2,k0+3)
//   B 4x16  : 2 VGPRs. lanes 0-15 -> N=lane, K rows (k0,k0+1); lanes 16-31 -> N=lane-16, rows (k0+2,k0+3)
//   C 16x16 : 8 VGPRs. VGPR r: lanes 0-15 -> (M=r, N=lane); lanes 16-31 -> (M=r+8, N=lane-16)
// Bp layout: [K/2][NCPAD][2]; element (k,col) at ((k>>1)*NCPAD + col)*2 + (k&1).
// M must be a multiple of 16 (50000 = 3125*16); cols padded to NCPAD with zeros.
template <int K, int NCPAD, int NCSTORE>
__global__ void __launch_bounds__(32)
gemm_wmma_f32_kernel(const float* __restrict__ A, const float* __restrict__ Bp,
                     float* __restrict__ C) {
  const int lane = threadIdx.x & 31;
  const int half = lane >> 4;
  const int l    = lane & 15;
  const int rowBase = blockIdx.x * 16;
  const int colBase = blockIdx.y * 16;
  const int col = colBase + l;

  v8f acc = {};
  const float* ap = A + (size_t)(rowBase + l) * K + 2 * half;      // 8B aligned
  const float* bp = Bp + ((size_t)half * NCPAD + col) * 2;         // 8B aligned

#pragma unroll
  for (int k0 = 0; k0 < K; k0 += 4) {
    v2f a = *(const v2f*)(ap + k0);                  // A[m][k0+2h .. k0+2h+1]
    v2f b = *(const v2f*)(bp + (k0 / 2) * NCPAD * 2); // B[k0+2h..+1][col]
    acc = __builtin_amdgcn_wmma_f32_16x16x4_f32(
        /*neg_a=*/false, a, /*neg_b=*/false, b,
        /*c_mod=*/(short)0, acc, /*reuse_a=*/false, /*reuse_b=*/false);
  }

  if (NCSTORE == NCPAD || col < NCSTORE) {
#pragma unroll
    for (int r = 0; r < 8; ++r) {
      C[(size_t)(rowBase + r + 8 * half) * NCSTORE + col] = acc[r];
    }
  }
}

// ---------------- pack kernels ---------------------------------------------

// W1[h][f][o] (4x128x32) -> Bp1 (K=128, NCPAD=128): col = h*32+o, k = f
__global__ void pack_W1_kernel(const float* __restrict__ W1, float* __restrict__ Bp) {
  int i = blockIdx.x * blockDim.x + threadIdx.x;
  if (i >= GAT_H * GAT_FIN * GAT_FH) return;
  int h   = i / (GAT_FIN * GAT_FH);
  int rem = i % (GAT_FIN * GAT_FH);
  int k   = rem / GAT_FH;          // f
  int o   = rem % GAT_FH;
  int col = h * GAT_FH + o;
  Bp[((size_t)(k >> 1) * GAT_C1 + col) * 2 + (k & 1)] = W1[i];
}

// W2[k][c] (128x40) -> Bp2 (K=128, NCPAD=48), zero-padded cols 40..47
__global__ void pack_W2_kernel(const float* __restrict__ W2, float* __restrict__ Bp) {
  int i = blockIdx.x * blockDim.x + threadIdx.x;
  if (i >= GAT_FIN * GAT_NCP) return;
  int k = i / GAT_NCP;
  int c = i % GAT_NCP;
  float v = (c < GAT_NC) ? W2[k * GAT_NC + c] : 0.0f;
  Bp[((size_t)(k >> 1) * GAT_NCP + c) * 2 + (k & 1)] = v;
}

// ---------------- small helper kernels -------------------------------------

__global__ void fill_kernel(float* __restrict__ p, float v, int n) {
  int i = blockIdx.x * blockDim.x + threadIdx.x;
  if (i < n) p[i] = v;
}

// ss[n,h] = proj[n,h,:].a_src[h,:] ; st likewise. proj layout [N, H*F].
__global__ void scores_kernel(const float* __restrict__ proj,
                              const float* __restrict__ a_src,
                              const float* __restrict__ a_tgt,
                              float* __restrict__ ss, float* __restrict__ st,
                              int N, int H, int F) {
  int i = blockIdx.x * blockDim.x + threadIdx.x;
  if (i >= N * H) return;
  int n = i / H, h = i % H;
  const float* p  = proj + (size_t)n * H * F + (size_t)h * F;
  const float* as = a_src + (size_t)h * F;
  const float* at = a_tgt + (size_t)h * F;
  float s0 = 0.0f, s1 = 0.0f;
  for (int o = 0; o < F; ++o) {
    float v = p[o];
    s0 = fmaf(v, as[o], s0);
    s1 = fmaf(v, at[o], s1);
  }
  ss[i] = s0;
  st[i] = s1;
}

__device__ inline void atomicMaxF32(float* addr, float val) {
  // sign-aware integer trick; buffer initialized to -inf, values are finite.
  if (val >= 0.0f) atomicMax((int*)addr, __float_as_int(val));
  else             atomicMin((unsigned int*)addr, (unsigned int)__float_as_int(val));
}

__device__ inline float leaky(float v) { return v > 0.0f ? v : GAT_ALPHA * v; }

// segment-max of leakyrelu(ss[src]+st[tgt]) into maxv[tgt]
__global__ void edge_max_kernel(const int* __restrict__ src, const int* __restrict__ tgt,
                                const float* __restrict__ ss, const float* __restrict__ st,
                                float* __restrict__ maxv, int E, int H) {
  int i = blockIdx.x * blockDim.x + threadIdx.x;
  if (i >= E * H) return;
  int e = i / H, h = i % H;
  int s = src[e], t = tgt[e];
  float v = leaky(ss[s * H + h] + st[t * H + h]);
  atomicMaxF32(&maxv[t * H + h], v);
}

// ex = exp(e - m[tgt]); store ex; denom[tgt] += ex
__global__ void edge_exp_kernel(const int* __restrict__ src, const int* __restrict__ tgt,
                                const float* __restrict__ ss, const float* __restrict__ st,
                                const float* __restrict__ maxv,
                                float* __restrict__ exbuf, float* __restrict__ denom,
                                int E, int H) {
  int i = blockIdx.x * blockDim.x + threadIdx.x;
  if (i >= E * H) return;
  int e = i / H, h = i % H;
  int s = src[e], t = tgt[e];
  float v = leaky(ss[s * H + h] + st[t * H + h]);
  float m = maxv[t * H + h];
  if (!(m > -INFINITY)) m = 0.0f;   // reference: non-finite max -> 0
  float ex = expf(v - m);
  exbuf[i] = ex;
  atomicAdd(&denom[t * H + h], ex);
}

// out[tgt, c] += proj[src, c] * exbuf[e,h] / (denom[tgt,h] + 1e-16)
__global__ void edge_aggr_kernel(const int* __restrict__ src, const int* __restrict__ tgt,
                                 const float* __restrict__ proj,
                                 const float* __restrict__ exbuf,
                                 const float* __restrict__ denom,
                                 float* __restrict__ out,
                                 int E, int H, int F) {
  const int C = H * F;
  long long i = (long long)blockIdx.x * blockDim.x + threadIdx.x;
  if (i >= (long long)E * C) return;
  int e = (int)(i / C);
  int c = (int)(i % C);
  int h = c / F;
  int s = src[e], t = tgt[e];
  float att = exbuf[(size_t)e * H + h] / (denom[t * H + h] + 1e-16f);
  atomicAdd(&out[(size_t)t * C + c], proj[(size_t)s * C + c] * att);
}

// in-place h = elu(h + b[c])
__global__ void bias_elu_kernel(float* __restrict__ h, const float* __restrict__ b,
                                int total, int C) {
  int i = blockIdx.x * blockDim.x + threadIdx.x;
  if (i >= total) return;
  float v = h[i] + b[i % C];
  h[i] = v > 0.0f ? v : (expf(v) - 1.0f);
}

__global__ void log_softmax_kernel(const float* __restrict__ in, float* __restrict__ out,
                                   int N, int C) {
  int n = blockIdx.x * blockDim.x + threadIdx.x;
  if (n >= N) return;
  const float* p = in + (size_t)n * C;
  float m = -INFINITY;
  for (int c = 0; c < C; ++c) m = fmaxf(m, p[c]);
  float s = 0.0f;
  for (int c = 0; c < C; ++c) s += expf(p[c] - m);
  float ls = m + logf(s);
  float* o = out + (size_t)n * C;
  for (int c = 0; c < C; ++c) o[c] = p[c] - ls;
}

// ---------------------------------------------------------------------------

extern "C" void kernel_launch(void* const* d_in, const int* in_sizes, int n_in,
                              void* d_out, int out_size, void* d_ws, size_t ws_size,
                              hipStream_t stream) {
  const float* x   = (const float*)d_in[0];
  // d_in[1] = adj placeholder (unused)
  const int*   ei  = (const int*)d_in[2];     // [2, E]
  const float* W1  = (const float*)d_in[3];   // [4,128,32]
  const float* a1s = (const float*)d_in[4];   // [4,32]
  const float* a1t = (const float*)d_in[5];   // [4,32]
  const float* b1  = (const float*)d_in[6];   // [128]
  const float* W2  = (const float*)d_in[7];   // [128,40]
  const float* a2s = (const float*)d_in[8];   // [40]
  const float* a2t = (const float*)d_in[9];   // [40]
  float* out = (float*)d_out;                  // [N,40]

  constexpr int N = GAT_N, E = GAT_E, H = GAT_H, FH = GAT_FH, C1 = GAT_C1, NC = GAT_NC;
  const int* src = ei;
  const int* tgt = ei + E;

  // ---- workspace carve-out (256B aligned) ----
  char*  ws  = (char*)d_ws;
  size_t off = 0;
  auto alloc = [&](size_t bytes) -> float* {
    float* p = (float*)(ws + off);
    off = (off + bytes + 255) & ~(size_t)255;
    return p;
  };
  float* Bp1  = alloc((size_t)GAT_FIN * C1 * 4);      //  64 KB  packed W1
  float* Bp2  = alloc((size_t)GAT_FIN * GAT_NCP * 4); //  24 KB  packed+padded W2
  float* proj = alloc((size_t)N * C1 * 4);            //  25.6 MB proj1, reused as proj2
  float* ss   = alloc((size_t)N * H * 4);
  float* st   = alloc((size_t)N * H * 4);
  float* maxv = alloc((size_t)N * H * 4);
  float* den  = alloc((size_t)N * H * 4);
  float* exb  = alloc((size_t)E * H * 4);             //  12.8 MB
  float* agg  = alloc((size_t)N * C1 * 4);            //  25.6 MB layer1 out -> h (in-place ELU)
  float* out2 = alloc((size_t)N * NC * 4);            //   8.0 MB pre-softmax logits
  (void)ws_size; (void)in_sizes; (void)n_in; (void)out_size;

  const int B256 = 256;

  // ================= layer 1 =================
  pack_W1_kernel<<<(H * GAT_FIN * FH + B256 - 1) / B256, B256, 0, stream>>>(W1, Bp1);
  pack_W2_kernel<<<(GAT_FIN * GAT_NCP + B256 - 1) / B256, B256, 0, stream>>>(W2, Bp2);

  gemm_wmma_f32_kernel<GAT_FIN, GAT_C1, GAT_C1>
      <<<dim3(N / 16, C1 / 16), 32, 0, stream>>>(x, Bp1, proj);

  scores_kernel<<<(N * H + B256 - 1) / B256, B256, 0, stream>>>(proj, a1s, a1t, ss, st, N, H, FH);

  fill_kernel<<<(N * H + B256 - 1) / B256, B256, 0, stream>>>(maxv, -INFINITY, N * H);
  fill_kernel<<<(N * H + B256 - 1) / B256, B256, 0, stream>>>(den, 0.0f, N * H);
  fill_kernel<<<(N * C1 + B256 - 1) / B256, B256, 0, stream>>>(agg, 0.0f, N * C1);

  edge_max_kernel<<<(E * H + B256 - 1) / B256, B256, 0, stream>>>(src, tgt, ss, st, maxv, E, H);
  edge_exp_kernel<<<(E * H + B256 - 1) / B256, B256, 0, stream>>>(src, tgt, ss, st, maxv, exb, den, E, H);
  {
    long long tot = (long long)E * C1;   // 102.4M (edge, channel) pairs
    edge_aggr_kernel<<<(unsigned)((tot + B256 - 1) / B256), B256, 0, stream>>>(
        src, tgt, proj, exb, den, agg, E, H, FH);
  }

  bias_elu_kernel<<<(N * C1 + B256 - 1) / B256, B256, 0, stream>>>(agg, b1, N * C1, C1);

  // ================= layer 2 =================
  gemm_wmma_f32_kernel<GAT_C1, GAT_NCP, GAT_NC>
      <<<dim3(N / 16, GAT_NCP / 16), 32, 0, stream>>>(agg, Bp2, proj);

  scores_kernel<<<(N + B256 - 1) / B256, B256, 0, stream>>>(proj, a2s, a2t, ss, st, N, 1, NC);

  fill_kernel<<<(N + B256 - 1) / B256, B256, 0, stream>>>(maxv, -INFINITY, N);
  fill_kernel<<<(N + B256 - 1) / B256, B256, 0, stream>>>(den, 0.0f, N);
  fill_kernel<<<(N * NC + B256 - 1) / B256, B256, 0, stream>>>(out2, 0.0f, N * NC);

  edge_max_kernel<<<(E + B256 - 1) / B256, B256, 0, stream>>>(src, tgt, ss, st, maxv, E, 1);
  edge_exp_kernel<<<(E + B256 - 1) / B256, B256, 0, stream>>>(src, tgt, ss, st, maxv, exb, den, E, 1);
  {
    long long tot = (long long)E * NC;   // 32M pairs
    edge_aggr_kernel<<<(unsigned)((tot + B256 - 1) / B256), B256, 0, stream>>>(
        src, tgt, proj, exb, den, out2, E, 1, NC);
  }

  log_softmax_kernel<<<(N + B256 - 1) / B256, B256, 0, stream>>>(out2, out, N, NC);
}